// FeatEmb_82042465289117
// MI455X (gfx1250) — compile-verified
//
#include <hip/hip_runtime.h>
#include <hip/hip_bf16.h>
#include <algorithm>

// ---------------------------------------------------------------------------
// Problem constants (from reference)
// ---------------------------------------------------------------------------
#define NPTS   15872
#define KNN_K  5

typedef float v2f __attribute__((ext_vector_type(2)));
typedef float v8f __attribute__((ext_vector_type(8)));

#define EPI_BN_LEAKY 0
#define EPI_BIAS     1

// ---------------------------------------------------------------------------
// WMMA f32 GEMM: C[M x Nc] = A[M x Kd] * B[Kd x Nc], row-major, f32.
// One wave per 16x32 C slab (two 16x16 WMMA tiles sharing the A fragments,
// giving two independent accumulation chains). K processed in chunks of 16:
// all fragments loaded first (clause + one wait), then 8 interleaved
// V_WMMA_F32_16X16X4_F32; guarded tail for Kd % 16 != 0.
// M % 16 == 0 and Nc % 32 == 0 guaranteed by all call sites.
// Epilogue: eval-mode BN (+LeakyReLU 0.2) folded to x*scale+shift, or +bias.
// ---------------------------------------------------------------------------
template <int EPI>
__global__ __launch_bounds__(32)
void gemm_wmma(const float* __restrict__ A, const float* __restrict__ B,
               float* __restrict__ C, int M, int Kd, int Nc,
               const float* __restrict__ bn_g, const float* __restrict__ bn_b,
               const float* __restrict__ bn_m, const float* __restrict__ bn_v,
               const float* __restrict__ bias) {
  const int lane = threadIdx.x;
  const int hi   = lane >> 4;    // 0: K pair {0,1}; 1: K pair {2,3}
  const int l15  = lane & 15;    // m for A, n for B
  const long r   = (long)blockIdx.x * 16 + l15;   // A row this lane feeds
  const int  c0  = blockIdx.y * 32 + l15;         // col, first tile
  const int  c1  = c0 + 16;                       // col, second tile

  v8f acc0, acc1;
#pragma unroll
  for (int q = 0; q < 8; ++q) { acc0[q] = 0.0f; acc1[q] = 0.0f; }

  const float* Arow = A + r * (long)Kd;
  const int Kmain = Kd & ~15;
  int kk = 0;
  for (; kk < Kmain; kk += 16) {
    v2f av[4], bv0[4], bv1[4];
#pragma unroll
    for (int u = 0; u < 4; ++u) {
      const int k0 = kk + 4 * u + hi * 2;
      av[u].x = Arow[k0];
      av[u].y = Arow[k0 + 1];
    }
#pragma unroll
    for (int u = 0; u < 4; ++u) {
      const int k0 = kk + 4 * u + hi * 2;
      bv0[u].x = B[(long)k0 * Nc + c0];
      bv0[u].y = B[(long)(k0 + 1) * Nc + c0];
      bv1[u].x = B[(long)k0 * Nc + c1];
      bv1[u].y = B[(long)(k0 + 1) * Nc + c1];
    }
    if (kk + 16 < Kd)  // pull next B chunk toward cache (global_prefetch_b8)
      __builtin_prefetch(&B[(long)(kk + 16 + hi * 2) * Nc + c0], 0, 1);
#pragma unroll
    for (int u = 0; u < 4; ++u) {
      acc0 = __builtin_amdgcn_wmma_f32_16x16x4_f32(false, av[u], false, bv0[u],
                                                   (short)0, acc0, false, false);
      acc1 = __builtin_amdgcn_wmma_f32_16x16x4_f32(false, av[u], false, bv1[u],
                                                   (short)0, acc1, false, false);
    }
  }
  // K tail (Kd = 70, 227): guarded, zero-padded
  for (; kk < Kd; kk += 4) {
    const int k0 = kk + hi * 2;
    v2f av, bv0, bv1;
    av.x  = (k0     < Kd) ? Arow[k0]     : 0.0f;
    av.y  = (k0 + 1 < Kd) ? Arow[k0 + 1] : 0.0f;
    bv0.x = (k0     < Kd) ? B[(long)k0 * Nc + c0]       : 0.0f;
    bv0.y = (k0 + 1 < Kd) ? B[(long)(k0 + 1) * Nc + c0] : 0.0f;
    bv1.x = (k0     < Kd) ? B[(long)k0 * Nc + c1]       : 0.0f;
    bv1.y = (k0 + 1 < Kd) ? B[(long)(k0 + 1) * Nc + c1] : 0.0f;
    acc0 = __builtin_amdgcn_wmma_f32_16x16x4_f32(false, av, false, bv0,
                                                 (short)0, acc0, false, false);
    acc1 = __builtin_amdgcn_wmma_f32_16x16x4_f32(false, av, false, bv1,
                                                 (short)0, acc1, false, false);
  }

  float scale0 = 1.0f, shift0 = 0.0f, scale1 = 1.0f, shift1 = 0.0f;
  if (EPI == EPI_BN_LEAKY) {
    scale0 = bn_g[c0] * rsqrtf(bn_v[c0] + 1e-5f);
    shift0 = bn_b[c0] - bn_m[c0] * scale0;
    scale1 = bn_g[c1] * rsqrtf(bn_v[c1] + 1e-5f);
    shift1 = bn_b[c1] - bn_m[c1] * scale1;
  } else {
    shift0 = bias[c0];
    shift1 = bias[c1];
  }
#pragma unroll
  for (int q = 0; q < 8; ++q) {  // D layout: row = q + hi*8, col = l15
    const long row = (long)blockIdx.x * 16 + q + hi * 8;
    float x0 = acc0[q] * scale0 + shift0;
    float x1 = acc1[q] * scale1 + shift1;
    if (EPI == EPI_BN_LEAKY) {
      x0 = (x0 >= 0.0f) ? x0 : 0.2f * x0;
      x1 = (x1 >= 0.0f) ? x1 : 0.2f * x1;
    }
    C[row * (long)Nc + c0] = x0;
    C[row * (long)Nc + c1] = x1;
  }
}

// ---------------------------------------------------------------------------
// Fused KNN (block 1) helpers
// ---------------------------------------------------------------------------
__device__ __forceinline__ void knn_loadB(const float* __restrict__ f0,
                                          long cc, int hi, v2f* breg) {
#pragma unroll
  for (int t = 0; t < 16; ++t) {
    const int k0 = 4 * t + hi * 2;
    breg[t] = *(const v2f*)(f0 + cc * 64 + k0);
  }
}

__device__ __forceinline__ void knn_tile(const v2f* areg, const v2f* breg,
                                         const float* __restrict__ xx,
                                         int c0, int lane, int hi, int l15,
                                         float* tile, float* kv, int* ki) {
  const float xxc = xx[c0 + l15];
  // two independent accumulation chains (even / odd K-steps)
  v8f acc_a, acc_b;
#pragma unroll
  for (int q = 0; q < 8; ++q) { acc_a[q] = 0.0f; acc_b[q] = 0.0f; }
#pragma unroll
  for (int t = 0; t < 16; t += 2) {
    acc_a = __builtin_amdgcn_wmma_f32_16x16x4_f32(false, areg[t], false, breg[t],
                                                  (short)0, acc_a, false, false);
    acc_b = __builtin_amdgcn_wmma_f32_16x16x4_f32(false, areg[t + 1], false,
                                                  breg[t + 1], (short)0, acc_b,
                                                  false, false);
  }
  __syncthreads();
#pragma unroll
  for (int q = 0; q < 8; ++q) {
    const int rl = q + hi * 8;
    tile[rl * 16 + l15] = 2.0f * (acc_a[q] + acc_b[q]) - xxc;
  }
  __syncthreads();
  if (lane < 16) {
    for (int cI = 0; cI < 16; ++cI) {
      const float s = tile[lane * 16 + cI];
      if (s > kv[5]) {
        float cv = s; int cidx = c0 + cI;
#pragma unroll
        for (int p = 0; p < 6; ++p) {
          if (cv > kv[p]) {
            float tv = kv[p]; int ti = ki[p];
            kv[p] = cv; ki[p] = cidx; cv = tv; cidx = ti;
          }
        }
      }
    }
  }
  __syncthreads();
}

// ---------------------------------------------------------------------------
// Fused KNN (block 1): one wave per 16-row panel; A panel (16x64) register-
// resident across all 992 column tiles; B panel double-buffered (bA/bB) so the
// next tile's 16 loads fly during the current tile's top-k update. Top-6 per
// row kept in registers of lanes 0..15 (ranking key 2*inner - xx[col] is a
// per-row monotone transform of neg_sq_dist; strict '>' insertion in ascending
// column order == top_k lower-index tie-break; rank 0 = self, dropped).
// ---------------------------------------------------------------------------
__global__ __launch_bounds__(32)
void knn_kernel(const float* __restrict__ f0, const float* __restrict__ xx,
                int* __restrict__ idx_out, int n_pts) {
  __shared__ float tile[256];
  const int lane = threadIdx.x;
  const int hi   = lane >> 4;
  const int l15  = lane & 15;
  const long r   = (long)blockIdx.x * 16 + l15;

  v2f areg[16];
#pragma unroll
  for (int t = 0; t < 16; ++t) {
    const int k0 = 4 * t + hi * 2;
    areg[t] = *(const v2f*)(f0 + r * 64 + k0);
  }

  float kv[6];
  int   ki[6];
#pragma unroll
  for (int p = 0; p < 6; ++p) { kv[p] = -3.4e38f; ki[p] = 0; }

  const int ntiles = n_pts / 16;  // 992 (even)
  v2f bA[16], bB[16];
  knn_loadB(f0, (long)l15, hi, bA);
  for (int ct = 0; ct < ntiles; ct += 2) {
    // issue next tile's loads before consuming the current buffer
    if (ct + 1 < ntiles) knn_loadB(f0, (long)(ct + 1) * 16 + l15, hi, bB);
    knn_tile(areg, bA, xx, ct * 16, lane, hi, l15, tile, kv, ki);
    if (ct + 1 < ntiles) {
      if (ct + 2 < ntiles) knn_loadB(f0, (long)(ct + 2) * 16 + l15, hi, bA);
      knn_tile(areg, bB, xx, (ct + 1) * 16, lane, hi, l15, tile, kv, ki);
    }
  }
  if (lane < 16) {
#pragma unroll
    for (int j = 0; j < KNN_K; ++j) idx_out[r * KNN_K + j] = ki[j + 1];
  }
}

// ---------------------------------------------------------------------------
// Elementwise / reduction helpers
// ---------------------------------------------------------------------------
__global__ __launch_bounds__(256)
void build_edge(const float* __restrict__ feat, const int* __restrict__ idx,
                float* __restrict__ E, int n_pts, int i_dim) {
  const long total = (long)n_pts * KNN_K * 2 * i_dim;
  const int  w = 2 * i_dim;
  for (long t = (long)blockIdx.x * blockDim.x + threadIdx.x; t < total;
       t += (long)gridDim.x * blockDim.x) {
    const int  c  = (int)(t % w);
    const long rj = t / w;
    const int  j  = (int)(rj % KNN_K);
    const long n  = rj / KNN_K;
    float v;
    if (c < i_dim) {
      const long nb = idx[n * KNN_K + j];
      v = feat[nb * i_dim + c] - feat[n * i_dim + c];
    } else {
      v = feat[n * i_dim + (c - i_dim)];
    }
    E[t] = v;
  }
}

__global__ __launch_bounds__(256)
void max_over_k(const float* __restrict__ H2, float* __restrict__ f1,
                int n_pts, int o) {
  const long t = (long)blockIdx.x * blockDim.x + threadIdx.x;
  if (t >= (long)n_pts * o) return;
  const int  c = (int)(t % o);
  const long n = t / o;
  const float* base = H2 + n * KNN_K * (long)o + c;
  float m = base[0];
#pragma unroll
  for (int j = 1; j < KNN_K; ++j) m = fmaxf(m, base[(long)j * o]);
  f1[t] = m;
}

__global__ void pool_stage1(const float* __restrict__ f1, float* __restrict__ pm,
                            float* __restrict__ ps, int n_pts, int o) {
  const int c = threadIdx.x;
  const int b = blockIdx.x;
  const int chunk = (n_pts + gridDim.x - 1) / gridDim.x;
  const int s = b * chunk;
  const int e = (s + chunk < n_pts) ? s + chunk : n_pts;
  float mx = -3.4e38f, sm = 0.0f;
  for (int n = s; n < e; ++n) {
    const float x = f1[(long)n * o + c];
    mx = fmaxf(mx, x);
    sm += x;
  }
  pm[b * o + c] = mx;
  ps[b * o + c] = sm;
}

__global__ void pool_stage2(const float* __restrict__ pm, const float* __restrict__ ps,
                            float* __restrict__ red, int nb, int o, int n_pts) {
  const int c = threadIdx.x;
  float mx = -3.4e38f, sm = 0.0f;
  for (int b = 0; b < nb; ++b) {
    mx = fmaxf(mx, pm[b * o + c]);
    sm += ps[b * o + c];
  }
  red[c]     = mx;
  red[o + c] = sm / (float)n_pts;
}

__global__ __launch_bounds__(256)
void concat_x(const float* __restrict__ feat, const float* __restrict__ f1,
              const float* __restrict__ red, float* __restrict__ X,
              int n_pts, int i_dim, int o) {
  const int  w = i_dim + 3 * o;
  const long total = (long)n_pts * w;
  for (long t = (long)blockIdx.x * blockDim.x + threadIdx.x; t < total;
       t += (long)gridDim.x * blockDim.x) {
    const int  c = (int)(t % w);
    const long n = t / w;
    float v;
    if (c < i_dim)            v = feat[n * i_dim + c];
    else if (c < i_dim + o)   v = f1[n * o + (c - i_dim)];
    else if (c < i_dim + 2*o) v = red[c - i_dim - o];        // global max
    else                      v = red[o + (c - i_dim - 2*o)]; // global mean
    X[t] = v;
  }
}

__global__ __launch_bounds__(256)
void row_sqnorm(const float* __restrict__ f0, float* __restrict__ xx,
                int n_pts, int c_dim) {
  const int n = blockIdx.x * blockDim.x + threadIdx.x;
  if (n >= n_pts) return;
  float s = 0.0f;
  for (int c = 0; c < c_dim; ++c) {
    const float x = f0[(long)n * c_dim + c];
    s += x * x;
  }
  xx[n] = s;
}

__global__ __launch_bounds__(256)
void concat_out(const float* __restrict__ f0, const float* __restrict__ f1b,
                float* __restrict__ out, int n_pts) {
  const long total = (long)n_pts * 192;
  for (long t = (long)blockIdx.x * blockDim.x + threadIdx.x; t < total;
       t += (long)gridDim.x * blockDim.x) {
    const int  c = (int)(t % 192);
    const long n = t / 192;
    out[t] = (c < 64) ? f0[n * 64 + c] : f1b[n * 128 + (c - 64)];
  }
}

// ---------------------------------------------------------------------------
// Host side
// ---------------------------------------------------------------------------
struct BlockParams {
  const float *ec_w1, *ec_g, *ec_b, *ec_m, *ec_v, *ec_w2, *ec_b2;
  const float *fc1_w, *f1g, *f1b, *f1m, *f1v;
  const float *fc2_w, *f2g, *f2b, *f2m, *f2v;
  const float *fc3_w, *f3g, *f3b, *f3m, *f3v;
  const float *fc4_w, *fc4_b;
};

static BlockParams load_bp(void* const* d_in, int base, bool sorted_keys) {
  BlockParams p;
  const float** q = (const float**)d_in;
  if (!sorted_keys) {
    // insertion order: ec_w1, ec_bn(g,b,m,v), ec_w2, ec_b2, fc1_w, fc1_bn,
    //                  fc2_w, fc2_bn, fc3_w, fc3_bn, fc4_w, fc4_b
    p.ec_w1 = q[base+0];  p.ec_g = q[base+1]; p.ec_b = q[base+2];
    p.ec_m  = q[base+3];  p.ec_v = q[base+4]; p.ec_w2 = q[base+5];
    p.ec_b2 = q[base+6];
    p.fc1_w = q[base+7];  p.f1g = q[base+8];  p.f1b = q[base+9];
    p.f1m   = q[base+10]; p.f1v = q[base+11];
    p.fc2_w = q[base+12]; p.f2g = q[base+13]; p.f2b = q[base+14];
    p.f2m   = q[base+15]; p.f2v = q[base+16];
    p.fc3_w = q[base+17]; p.f3g = q[base+18]; p.f3b = q[base+19];
    p.f3m   = q[base+20]; p.f3v = q[base+21];
    p.fc4_w = q[base+22]; p.fc4_b = q[base+23];
  } else {
    // jax tree_flatten (sorted keys): ec_b2, ec_bn, ec_w1, ec_w2, fc1_bn,
    //                  fc1_w, fc2_bn, fc2_w, fc3_bn, fc3_w, fc4_b, fc4_w
    p.ec_b2 = q[base+0];
    p.ec_g  = q[base+1];  p.ec_b = q[base+2]; p.ec_m = q[base+3];
    p.ec_v  = q[base+4];
    p.ec_w1 = q[base+5];  p.ec_w2 = q[base+6];
    p.f1g   = q[base+7];  p.f1b = q[base+8];  p.f1m = q[base+9];
    p.f1v   = q[base+10]; p.fc1_w = q[base+11];
    p.f2g   = q[base+12]; p.f2b = q[base+13]; p.f2m = q[base+14];
    p.f2v   = q[base+15]; p.fc2_w = q[base+16];
    p.f3g   = q[base+17]; p.f3b = q[base+18]; p.f3m = q[base+19];
    p.f3v   = q[base+20]; p.fc3_w = q[base+21];
    p.fc4_b = q[base+22]; p.fc4_w = q[base+23];
  }
  return p;
}

static void run_block(const float* fin, int i_dim, int o_dim, const int* idx,
                      const BlockParams& p, float* blk_out,
                      float* bufE, float* bufH, float* pool,
                      float* pm, float* ps, float* red, hipStream_t stream) {
  const int M5 = NPTS * KNN_K;  // 79360, divisible by 16
  // 1) graph feature E[M5 x 2i]
  {
    long total = (long)M5 * 2 * i_dim;
    int nb = (int)std::min<long>((total + 255) / 256, 4096);
    build_edge<<<nb, 256, 0, stream>>>(fin, idx, bufE, NPTS, i_dim);
  }
  // 2) H1 = leaky(bn(E @ ec_w1)) : [M5 x 2o]
  gemm_wmma<EPI_BN_LEAKY><<<dim3(M5 / 16, (2 * o_dim) / 32), 32, 0, stream>>>(
      bufE, p.ec_w1, bufH, M5, 2 * i_dim, 2 * o_dim,
      p.ec_g, p.ec_b, p.ec_m, p.ec_v, nullptr);
  // 3) H2 = H1 @ ec_w2 + ec_b2 : [M5 x o]  (into bufE, E is dead)
  gemm_wmma<EPI_BIAS><<<dim3(M5 / 16, o_dim / 32), 32, 0, stream>>>(
      bufH, p.ec_w2, bufE, M5, 2 * o_dim, o_dim,
      nullptr, nullptr, nullptr, nullptr, p.ec_b2);
  // 4) f1 = max over k
  {
    long total = (long)NPTS * o_dim;
    max_over_k<<<(int)((total + 255) / 256), 256, 0, stream>>>(bufE, pool, NPTS, o_dim);
  }
  // 5) global max / mean over N
  pool_stage1<<<32, o_dim, 0, stream>>>(pool, pm, ps, NPTS, o_dim);
  pool_stage2<<<1, o_dim, 0, stream>>>(pm, ps, red, 32, o_dim, NPTS);
  // 6) X = [feat, f1, f1_max, f1_avg] : [N x (i+3o)]  (into bufH, H1 dead)
  {
    long total = (long)NPTS * (i_dim + 3 * o_dim);
    int nb = (int)std::min<long>((total + 255) / 256, 4096);
    concat_x<<<nb, 256, 0, stream>>>(fin, pool, red, bufH, NPTS, i_dim, o_dim);
  }
  // 7) Y1 = leaky(bn(X @ fc1_w)) : [N x 3o]   (into bufE)
  gemm_wmma<EPI_BN_LEAKY><<<dim3(NPTS / 16, (3 * o_dim) / 32), 32, 0, stream>>>(
      bufH, p.fc1_w, bufE, NPTS, i_dim + 3 * o_dim, 3 * o_dim,
      p.f1g, p.f1b, p.f1m, p.f1v, nullptr);
  // 8) Y2 = leaky(bn(Y1 @ fc2_w)) : [N x 2o]  (into bufH)
  gemm_wmma<EPI_BN_LEAKY><<<dim3(NPTS / 16, (2 * o_dim) / 32), 32, 0, stream>>>(
      bufE, p.fc2_w, bufH, NPTS, 3 * o_dim, 2 * o_dim,
      p.f2g, p.f2b, p.f2m, p.f2v, nullptr);
  // 9) Y3 = leaky(bn(Y2 @ fc3_w)) : [N x o]   (into bufE)
  gemm_wmma<EPI_BN_LEAKY><<<dim3(NPTS / 16, o_dim / 32), 32, 0, stream>>>(
      bufH, p.fc3_w, bufE, NPTS, 2 * o_dim, o_dim,
      p.f3g, p.f3b, p.f3m, p.f3v, nullptr);
  // 10) out = Y3 @ fc4_w + fc4_b : [N x o]
  gemm_wmma<EPI_BIAS><<<dim3(NPTS / 16, o_dim / 32), 32, 0, stream>>>(
      bufE, p.fc4_w, blk_out, NPTS, o_dim, o_dim,
      nullptr, nullptr, nullptr, nullptr, p.fc4_b);
}

extern "C" void kernel_launch(void* const* d_in, const int* in_sizes, int n_in,
                              void* d_out, int out_size, void* d_ws, size_t ws_size,
                              hipStream_t stream) {
  if (n_in < 50) return;
  const float* feat = (const float*)d_in[0];
  const int*   idx0 = (const int*)d_in[1];
  // detect flattening convention: insertion order -> d_in[2] is ec_w1 (70*128);
  // sorted-key order -> d_in[2] is ec_b2 (64).
  const bool sorted_keys = (in_sizes[2] == 64);
  BlockParams b0 = load_bp(d_in, 2, sorted_keys);
  BlockParams b1 = load_bp(d_in, 26, sorted_keys);

  // workspace layout (floats); requires ws_size >= ~143 MB
  float* wsf = (float*)d_ws;
  size_t off = 0;
  float* f0   = wsf + off; off += (size_t)NPTS * 64;
  float* f1b  = wsf + off; off += (size_t)NPTS * 128;
  float* xx   = wsf + off; off += (size_t)NPTS;
  int*   idx1 = (int*)(wsf + off); off += (size_t)NPTS * KNN_K;
  float* pool = wsf + off; off += (size_t)NPTS * 128;
  float* pm   = wsf + off; off += 32 * 128;
  float* ps   = wsf + off; off += 32 * 128;
  float* red  = wsf + off; off += 2 * 128;
  float* bufE = wsf + off; off += (size_t)NPTS * KNN_K * 128;  // E / H2 / Y1 / Y3
  float* bufH = wsf + off; off += (size_t)NPTS * KNN_K * 256;  // H1 / X / Y2
  if (ws_size < off * sizeof(float)) return;

  // ---- block 0: in 35 -> out 64, neighbor indices given ----
  run_block(feat, 35, 64, idx0, b0, f0, bufE, bufH, pool, pm, ps, red, stream);

  // ---- KNN on f0 (fused WMMA scoring + register top-6) ----
  row_sqnorm<<<(NPTS + 255) / 256, 256, 0, stream>>>(f0, xx, NPTS, 64);
  knn_kernel<<<NPTS / 16, 32, 0, stream>>>(f0, xx, idx1, NPTS);

  // ---- block 1: in 64 -> out 128 ----
  run_block(f0, 64, 128, idx1, b1, f1b, bufE, bufH, pool, pm, ps, red, stream);

  // ---- output: concat [f0, f1b] -> [N x 192] ----
  {
    long total = (long)NPTS * 192;
    int nb = (int)std::min<long>((total + 255) / 256, 8192);
    concat_out<<<nb, 256, 0, stream>>>(f0, f1b, (float*)d_out, NPTS);
  }
  (void)out_size;
}